// LaplacianLoss_78615081386501
// MI455X (gfx1250) — compile-verified
//
#include <hip/hip_runtime.h>

#define B_   16
#define N_   100000
#define K_   8
#define C_   3
#define TOT  (B_ * N_ * C_)        // 4,800,000 elements
#define NC   (N_ * C_)             // 300,000

typedef __attribute__((ext_vector_type(2))) float v2f;
typedef __attribute__((ext_vector_type(8))) float v8f;

// ---------------------------------------------------------------------------
// Wave-level reduction via V_WMMA_F32_16X16X4_F32.
// A (16x4 f32, 2 VGPRs): VGPR0 = lane partial (K=0 for lanes 0-15, K=2 for
// lanes 16-31), VGPR1 = 0.  B = all-ones (layout independent).  C = 0.
// D[m][n] = s_m + s_{m+16} for every column n.  Each lane sums its 8 C/D
// VGPRs: lanes 0-15 get sum(rows 0..7) = s_0..7 + s_16..23, lanes 16-31 get
// sum(rows 8..15) = s_8..15 + s_24..31.  Wave total = lane0 + lane16 values.
// Requires EXEC all-ones: call only from fully-converged code.
// ---------------------------------------------------------------------------
__device__ __forceinline__ float wave_reduce_wmma(float acc) {
    v2f a; a.x = acc;  a.y = 0.0f;
    v2f b; b.x = 1.0f; b.y = 1.0f;
    v8f c = {};
    c = __builtin_amdgcn_wmma_f32_16x16x4_f32(
        /*neg_a=*/false, a, /*neg_b=*/false, b,
        /*c_mod=*/(short)0, c, /*reuse_a=*/false, /*reuse_b=*/false);
    return c[0] + c[1] + c[2] + c[3] + c[4] + c[5] + c[6] + c[7];
}

__device__ __forceinline__ void block_epilogue(float acc, double* gacc, float* fout) {
    __shared__ float bsum;
    if (threadIdx.x == 0) bsum = 0.0f;
    __syncthreads();                         // reconverge -> EXEC all-ones
    float s = wave_reduce_wmma(acc);
    unsigned lane = threadIdx.x & 31u;
    if (lane == 0u || lane == 16u) atomicAdd(&bsum, s);   // ds_add_f32
    __syncthreads();
    if (threadIdx.x == 0) {
        if (gacc) atomicAdd(gacc, (double)bsum);          // global_atomic_add_f64
        else      atomicAdd(fout, bsum * (1.0f / (float)TOT));
    }
}

// ---------------------------------------------------------------------------
// Pass 1 (when workspace allows): diff = preds - targets, float4-vectorized.
// ---------------------------------------------------------------------------
__global__ __launch_bounds__(256)
void diff_kernel(const float4* __restrict__ P, const float4* __restrict__ T,
                 float4* __restrict__ D) {
    int stride = gridDim.x * blockDim.x;
    for (int i = blockIdx.x * blockDim.x + threadIdx.x; i < TOT / 4; i += stride) {
        float4 p = P[i], t = T[i];
        D[i] = make_float4(p.x - t.x, p.y - t.y, p.z - t.z, p.w - t.w);
    }
}

// ---------------------------------------------------------------------------
// Laplacian + squared-error + reduction.  One thread per n: loads the idx /
// weight row once, reuses it across all 16 batches.  FUSED=true computes the
// diff on the fly (2x gather loads); FUSED=false reads the precomputed diff.
// ---------------------------------------------------------------------------
template <bool FUSED>
__global__ __launch_bounds__(256)
void lap_kernel(const float* __restrict__ P, const float* __restrict__ T,
                const float* __restrict__ DF,
                const long long* __restrict__ idx, const float* __restrict__ w,
                double* gacc, float* fout) {
    float acc = 0.0f;
    int stride = gridDim.x * blockDim.x;
    for (int n = blockIdx.x * blockDim.x + threadIdx.x; n < N_; n += stride) {
        int   j[K_];
        float wk[K_];
#pragma unroll
        for (int k = 0; k < K_; ++k) {
            j[k]  = 3 * (int)idx[n * K_ + k];
            wk[k] = w[n * K_ + k];
        }
        const int n3 = 3 * n;
#pragma unroll 2
        for (int b = 0; b < B_; ++b) {
            const int bb = b * NC;
            float d0, d1, d2;
            if (FUSED) {
                const float* p = P + bb + n3;
                const float* t = T + bb + n3;
                d0 = p[0] - t[0]; d1 = p[1] - t[1]; d2 = p[2] - t[2];
#pragma unroll
                for (int k = 0; k < K_; ++k) {
                    const float* pg = P + bb + j[k];
                    const float* tg = T + bb + j[k];
                    d0 += wk[k] * (pg[0] - tg[0]);
                    d1 += wk[k] * (pg[1] - tg[1]);
                    d2 += wk[k] * (pg[2] - tg[2]);
                }
            } else {
                const float* f = DF + bb + n3;
                d0 = f[0]; d1 = f[1]; d2 = f[2];
#pragma unroll
                for (int k = 0; k < K_; ++k) {
                    const float* g = DF + bb + j[k];
                    d0 += wk[k] * g[0];
                    d1 += wk[k] * g[1];
                    d2 += wk[k] * g[2];
                }
            }
            acc += d0 * d0 + d1 * d1 + d2 * d2;
        }
    }
    block_epilogue(acc, gacc, fout);
}

__global__ void finalize_kernel(const double* gacc, float* out) {
    if (threadIdx.x == 0 && blockIdx.x == 0)
        out[0] = (float)(gacc[0] * (1.0 / (double)TOT));
}

// ---------------------------------------------------------------------------
extern "C" void kernel_launch(void* const* d_in, const int* in_sizes, int n_in,
                              void* d_out, int out_size, void* d_ws, size_t ws_size,
                              hipStream_t stream) {
    const float*     preds   = (const float*)d_in[0];
    const float*     targets = (const float*)d_in[1];
    const long long* nbs_idx = (const long long*)d_in[2];
    const float*     nbs_w   = (const float*)d_in[3];
    float*           out     = (float*)d_out;

    const int lap_blocks  = (N_ + 255) / 256;          // 391
    const int diff_blocks = (TOT / 4 + 255) / 256;     // 4688
    const size_t need_two_pass = 16 + (size_t)TOT * sizeof(float);

    if (ws_size >= 16) {
        double* gacc = (double*)d_ws;
        hipMemsetAsync(d_ws, 0, 16, stream);
        if (ws_size >= need_two_pass) {
            float* df = (float*)((char*)d_ws + 16);
            diff_kernel<<<diff_blocks, 256, 0, stream>>>(
                (const float4*)preds, (const float4*)targets, (float4*)df);
            lap_kernel<false><<<lap_blocks, 256, 0, stream>>>(
                nullptr, nullptr, df, nbs_idx, nbs_w, gacc, nullptr);
        } else {
            lap_kernel<true><<<lap_blocks, 256, 0, stream>>>(
                preds, targets, nullptr, nbs_idx, nbs_w, gacc, nullptr);
        }
        finalize_kernel<<<1, 64, 0, stream>>>(gacc, out);
    } else {
        // No usable workspace: accumulate scaled float partials into d_out.
        hipMemsetAsync(d_out, 0, sizeof(float), stream);
        lap_kernel<true><<<lap_blocks, 256, 0, stream>>>(
            preds, targets, nullptr, nbs_idx, nbs_w, nullptr, out);
    }
}